// KalmanFilter_2886218023160
// MI455X (gfx1250) — compile-verified
//
#include <hip/hip_runtime.h>
#include <hip/hip_bf16.h>
#include <math.h>

// ---------------------------------------------------------------------------
// Batched Kalman filter scan: 4096 independent 4-state filters, 2048 steps.
// One filter per lane; symmetric covariance (10 f32) + mean (4 f32) in VGPRs.
// Data pipeline: 12-step-deep async ring buffer in LDS using CDNA5
// GLOBAL_LOAD_ASYNC_TO_LDS_B32 (ASYNCcnt, th:NT since input is stream-once),
// consumed with in-order s_wait_asynccnt; s_wait_dscnt 0 guards the WAR on
// slot recycling. Output stored non-temporally (write-once, never re-read).
// 128 single-wave workgroups spread the (hard-capped) 128 waves over WGPs.
// ---------------------------------------------------------------------------

#define T_STEPS 2048
#define BATCH_N 4096
#define BLOCK   32         // one wave per workgroup (wave32)
#define DEPTH   12         // async pipeline depth: 5*12=60 <= 63 (6-bit ASYNCcnt)

typedef __attribute__((ext_vector_type(2))) float v2f;
typedef __attribute__((ext_vector_type(4))) float v4f;

// Issue 5 async dword copies global->LDS for one lane's (z0,z1,L0,L1,L2).
// INST_OFFSET applies to both the LDS and global addresses, so one base each.
// NT hint: input is streamed exactly once; don't retain in caches.
__device__ __forceinline__ void async_load5(unsigned lds_off, const float* g) {
    asm volatile(
        "global_load_async_to_lds_b32 %0, %1, off th:TH_LOAD_NT\n\t"
        "global_load_async_to_lds_b32 %0, %1, off offset:4 th:TH_LOAD_NT\n\t"
        "global_load_async_to_lds_b32 %0, %1, off offset:8 th:TH_LOAD_NT\n\t"
        "global_load_async_to_lds_b32 %0, %1, off offset:12 th:TH_LOAD_NT\n\t"
        "global_load_async_to_lds_b32 %0, %1, off offset:16 th:TH_LOAD_NT"
        :: "v"(lds_off), "v"(g) : "memory");
}

__global__ __launch_bounds__(BLOCK, 1)
void kf_scan_kernel(const float* __restrict__ z_and_L,   // [T, N, 5]
                    const v4f*   __restrict__ mu0,       // [N, 4]
                    v2f*         __restrict__ y_out) {   // [T, N, 2]
    const int tid = threadIdx.x;
    const int n   = blockIdx.x * BLOCK + tid;

    // Ring buffer: DEPTH slots x 32 lanes x 5 dwords. Lane stride = 5 dwords
    // -> gcd(5,64)=1, conflict-free ds_load_b32 across the wave.
    __shared__ float ring[DEPTH * BLOCK * 5];

    // Low 32 bits of a generic pointer into LDS are the LDS byte offset.
    const unsigned lds_base =
        (unsigned)(unsigned long long)(uintptr_t)&ring[tid * 5];
    const float* gz = z_and_L + (size_t)n * 5;

    // ---- prologue: fill the pipeline ------------------------------------
    #pragma unroll
    for (int s = 0; s < DEPTH; ++s)
        async_load5(lds_base + (unsigned)(s * BLOCK * 5 * 4),
                    gz + (size_t)s * BATCH_N * 5);

    // ---- state init -----------------------------------------------------
    v4f m = mu0[n];
    float m0 = m.x, m1 = m.y, m2 = m.z, m3 = m.w;
    float s00 = 1.f, s01 = 0.f, s02 = 0.f, s03 = 0.f;
    float s11 = 1.f, s12 = 0.f, s13 = 0.f;
    float s22 = 1.f, s23 = 0.f;
    float s33 = 1.f;

    // A = I + DT*[[-b/m,0,-k/m,0],[0,-b/m,0,-k/m],[1,0,0,0],[0,1,0,0]]
    constexpr float Aa = 1.0f + (-0.1f / 1.0f) * 0.01f;  // 0.999
    constexpr float Ac = (-1.0f / 1.0f) * 0.01f;          // -0.01
    constexpr float Ad = 0.01f;                           // DT

    for (int t = 0; t < T_STEPS; ++t) {
        // Async loads complete in order: <=5*(DEPTH-1) leaves the newest
        // DEPTH-1 steps outstanding => the oldest (step t) has landed in LDS.
        asm volatile("s_wait_asynccnt 55" ::: "memory");

        const int slot = t % DEPTH;
        const float* l = &ring[(slot * BLOCK + tid) * 5];
        const float z0 = l[0], z1 = l[1];
        const float L0 = l[2], L1 = l[3], L2 = l[4];

        // DS reads of this slot must retire before the async write recycles it.
        asm volatile("s_wait_dscnt 0" ::: "memory");

        // Refill this slot with step t+DEPTH (clamped in the tail so the
        // wait threshold above stays valid; dead slots are never re-read).
        int ts = t + DEPTH; if (ts > T_STEPS - 1) ts = T_STEPS - 1;
        async_load5(lds_base + (unsigned)(slot * BLOCK * 5 * 4),
                    gz + (size_t)ts * BATCH_N * 5);

        // ---- predict: mu_pred = A mu ------------------------------------
        const float n0 = fmaf(Ac, m2, Aa * m0);
        const float n1 = fmaf(Ac, m3, Aa * m1);
        const float n2 = fmaf(Ad, m0, m2);
        const float n3 = fmaf(Ad, m1, m3);

        // ---- Tq = A * Sig (Sig symmetric), A rows have 2 non-zeros ------
        const float t00 = fmaf(Ac, s02, Aa * s00);
        const float t01 = fmaf(Ac, s12, Aa * s01);
        const float t02 = fmaf(Ac, s22, Aa * s02);
        const float t03 = fmaf(Ac, s23, Aa * s03);
        const float t10 = fmaf(Ac, s03, Aa * s01);
        const float t11 = fmaf(Ac, s13, Aa * s11);
        const float t12 = fmaf(Ac, s23, Aa * s12);
        const float t13 = fmaf(Ac, s33, Aa * s13);
        const float t20 = fmaf(Ad, s00, s02);
        const float t21 = fmaf(Ad, s01, s12);
        const float t22 = fmaf(Ad, s02, s22);
        const float t23 = fmaf(Ad, s03, s23);
        const float t31 = fmaf(Ad, s11, s13);

        // ---- Sig_pred = Tq * A^T + diag(1,1,0,0), upper triangle --------
        const float p00 = fmaf(Ac, t02, Aa * t00) + 1.0f;
        const float p01 = fmaf(Ac, t03, Aa * t01);
        const float p02 = fmaf(Ad, t00, t02);
        const float p03 = fmaf(Ad, t01, t03);
        const float p11 = fmaf(Ac, t13, Aa * t11) + 1.0f;
        const float p12 = fmaf(Ad, t10, t12);
        const float p13 = fmaf(Ad, t11, t13);
        const float p22 = fmaf(Ad, t20, t22);
        const float p23 = fmaf(Ad, t21, t23);
        const float p33 = fmaf(Ad, t31, fmaf(Ad, s13, s33));

        // ---- R from Cholesky params -------------------------------------
        const float l00 = __expf(L0);
        const float l10 = L1;
        const float l11 = __expf(L2);
        const float R00 = l00 * l00;
        const float R01 = l00 * l10;
        const float R11 = fmaf(l10, l10, l11 * l11);

        // ---- S = Sig_pred[2:4,2:4] + R ; 2x2 inverse --------------------
        const float S00 = p22 + R00;
        const float S01 = p23 + R01;
        const float S11 = p33 + R11;
        const float det  = fmaf(S00, S11, -(S01 * S01));
        const float idet = __builtin_amdgcn_rcpf(det);
        const float i00 =  S11 * idet;
        const float i01 = -(S01 * idet);
        const float i11 =  S00 * idet;

        // ---- K = Sig_pred[:,2:4] * Sinv (4x2) ---------------------------
        const float K00 = fmaf(p02, i00, p03 * i01);
        const float K01 = fmaf(p02, i01, p03 * i11);
        const float K10 = fmaf(p12, i00, p13 * i01);
        const float K11 = fmaf(p12, i01, p13 * i11);
        const float K20 = fmaf(p22, i00, p23 * i01);
        const float K21 = fmaf(p22, i01, p23 * i11);
        const float K30 = fmaf(p23, i00, p33 * i01);
        const float K31 = fmaf(p23, i01, p33 * i11);

        // ---- mean update ------------------------------------------------
        const float e0 = z0 - n2;
        const float e1 = z1 - n3;
        m0 = fmaf(K01, e1, fmaf(K00, e0, n0));
        m1 = fmaf(K11, e1, fmaf(K10, e0, n1));
        m2 = fmaf(K21, e1, fmaf(K20, e0, n2));
        m3 = fmaf(K31, e1, fmaf(K30, e0, n3));

        // ---- covariance update: Sig - K * (C Sig_pred), symmetric -------
        s00 = p00 - fmaf(K00, p02, K01 * p03);
        s01 = p01 - fmaf(K00, p12, K01 * p13);
        s02 = p02 - fmaf(K00, p22, K01 * p23);
        s03 = p03 - fmaf(K00, p23, K01 * p33);
        s11 = p11 - fmaf(K10, p12, K11 * p13);
        s12 = p12 - fmaf(K10, p22, K11 * p23);
        s13 = p13 - fmaf(K10, p23, K11 * p33);
        s22 = p22 - fmaf(K20, p22, K21 * p23);
        s23 = p23 - fmaf(K20, p23, K21 * p33);
        s33 = p33 - fmaf(K30, p23, K31 * p33);

        // ---- emit y = mu_out[2:4]; NT b64 store (write-once stream) -----
        v2f y; y.x = m2; y.y = m3;
        __builtin_nontemporal_store(y, &y_out[(size_t)t * BATCH_N + n]);
    }
}

extern "C" void kernel_launch(void* const* d_in, const int* in_sizes, int n_in,
                              void* d_out, int out_size, void* d_ws, size_t ws_size,
                              hipStream_t stream) {
    (void)in_sizes; (void)n_in; (void)out_size; (void)d_ws; (void)ws_size;
    const float* z   = (const float*)d_in[0];   // [2048, 4096, 5] f32
    const v4f*   mu0 = (const v4f*)d_in[1];     // [4096, 4] f32 (16B aligned)
    v2f*         y   = (v2f*)d_out;             // [2048, 4096, 2] f32

    dim3 grid(BATCH_N / BLOCK);   // 128 single-wave workgroups
    dim3 block(BLOCK);            // 32 threads = 1 wave32
    kf_scan_kernel<<<grid, block, 0, stream>>>(z, mu0, y);
}